// RGCNLowMemConv_8409545965796
// MI455X (gfx1250) — compile-verified
//
#include <hip/hip_runtime.h>

#define FEAT 64
#define NUM_RELS 8

typedef float v2f __attribute__((ext_vector_type(2)));
typedef float v8f __attribute__((ext_vector_type(8)));

// ---------------------------------------------------------------------------
// Zero the output (scatter kernels accumulate with atomics).
// ---------------------------------------------------------------------------
__global__ void rgcn_zero_kernel(float4* __restrict__ out, int n4) {
  int i = blockIdx.x * blockDim.x + threadIdx.x;
  if (i < n4) out[i] = make_float4(0.f, 0.f, 0.f, 0.f);
}

// ---------------------------------------------------------------------------
// Phase 1: T[rloc, n, o] = feat[n, :] @ weight[r0+rloc, :, :]
// Full-precision f32 tensor path: V_WMMA_F32_16X16X4_F32, K stepped 4 at a
// time, 16 WMMAs fully unrolled per 16x16 output tile.
// Block = 128 threads = 4 waves; wave w computes o-tile [w*16, w*16+16).
// blockIdx.x = node tile (16 rows), blockIdx.y = relation within chunk.
//
// A (16x4 f32, ISA 7.12.2): lanes 0-15 -> M=lane, v0=K0, v1=K1;
//                           lanes 16-31 -> M=lane-16, v0=K2, v1=K3.
// B (4x16 f32, transposed analog): lane%16 = N col; lanes<16 v0=K0,v1=K1;
//                                  lanes>=16 v0=K2,v1=K3.
// C/D (16x16 f32): VGPR v holds M = v + 8*(lane>=16), N = lane%16.
// ---------------------------------------------------------------------------
__global__ __launch_bounds__(128) void rgcn_gemm_wmma(
    const float* __restrict__ feat, const float* __restrict__ weight,
    float* __restrict__ T, int r0, int N) {
  const int node0 = blockIdx.x << 4;
  const int rloc  = blockIdx.y;
  const int wave  = threadIdx.x >> 5;
  const int lane  = threadIdx.x & 31;
  const int m     = lane & 15;       // row (A) / col (B/C)
  const int khalf = lane >> 4;       // which K pair this half-wave holds
  const int o0    = wave << 4;

  const float* __restrict__ Wr = weight + (size_t)(r0 + rloc) * (FEAT * FEAT);
  const float* __restrict__ ap = feat + (size_t)(node0 + m) * FEAT + (khalf << 1);
  const float* __restrict__ bp = Wr + ((khalf << 1) * FEAT) + o0 + m;

  v8f c = {};
#pragma unroll
  for (int k0 = 0; k0 < FEAT; k0 += 4) {
    v2f a, b;
    a.x = ap[k0];                 // A[m][k0 + 2*khalf + 0]
    a.y = ap[k0 + 1];             // A[m][k0 + 2*khalf + 1]
    b.x = bp[k0 * FEAT];          // B[k0 + 2*khalf + 0][o0+m]
    b.y = bp[k0 * FEAT + FEAT];   // B[k0 + 2*khalf + 1][o0+m]
    c = __builtin_amdgcn_wmma_f32_16x16x4_f32(
        /*neg_a=*/false, a, /*neg_b=*/false, b,
        /*c_mod=*/(short)0, c, /*reuse_a=*/false, /*reuse_b=*/false);
  }

  float* __restrict__ op = T + ((size_t)rloc * N + node0) * FEAT + o0 + m;
#pragma unroll
  for (int v = 0; v < 8; ++v) {
    op[(size_t)(v + (khalf << 3)) * FEAT] = c[v];  // row M = v + 8*khalf
  }
}

// ---------------------------------------------------------------------------
// Phase 2: per-edge gather from L2-resident slab, scale by norm, atomic
// scatter-add into out. 16 threads per edge, float4 per thread.
// ---------------------------------------------------------------------------
__global__ __launch_bounds__(256) void rgcn_edge_scatter(
    const float* __restrict__ T, const float* __restrict__ norm,
    const int* __restrict__ etypes, const int* __restrict__ src,
    const int* __restrict__ dst, float* __restrict__ out,
    int r0, int r1, int N, int E) {
  long long t = (long long)blockIdx.x * blockDim.x + threadIdx.x;
  int e = (int)(t >> 4);
  int j = (int)(t & 15);
  if (e >= E) return;
  int r = etypes[e];
  if (r < r0 || r >= r1) return;

  float nw = norm[e];
  const float4* __restrict__ row = reinterpret_cast<const float4*>(
      T + (((size_t)(r - r0) * N + (size_t)src[e]) << 6));
  float4 v = row[j];
  float* op = out + ((size_t)dst[e] << 6) + (j << 2);
  unsafeAtomicAdd(op + 0, v.x * nw);
  unsafeAtomicAdd(op + 1, v.y * nw);
  unsafeAtomicAdd(op + 2, v.z * nw);
  unsafeAtomicAdd(op + 3, v.w * nw);
}

// ---------------------------------------------------------------------------
// Fallback (workspace too small for even one relation slab): edge-direct
// vec-mat per edge via wave shuffles, no workspace needed.
// ---------------------------------------------------------------------------
__global__ __launch_bounds__(256) void rgcn_edge_direct(
    const float* __restrict__ feat, const float* __restrict__ norm,
    const int* __restrict__ etypes, const int* __restrict__ src,
    const int* __restrict__ dst, const float* __restrict__ weight,
    float* __restrict__ out, int E) {
  int wave = threadIdx.x >> 5;
  int lane = threadIdx.x & 31;
  long long e = (long long)blockIdx.x * 8 + wave;
  if (e >= E) return;

  const float* f = feat + ((size_t)src[e] << 6);
  float x0 = f[lane], x1 = f[lane + 32];
  const float* Wr = weight + (size_t)etypes[e] * (FEAT * FEAT);
  float acc0 = 0.f, acc1 = 0.f;
#pragma unroll
  for (int i = 0; i < 32; ++i) {
    float fi = __shfl(x0, i, 32);
    acc0 += fi * Wr[i * FEAT + lane];
    acc1 += fi * Wr[i * FEAT + lane + 32];
  }
#pragma unroll
  for (int i = 0; i < 32; ++i) {
    float fi = __shfl(x1, i, 32);
    acc0 += fi * Wr[(i + 32) * FEAT + lane];
    acc1 += fi * Wr[(i + 32) * FEAT + lane + 32];
  }
  float nw = norm[e];
  float* op = out + ((size_t)dst[e] << 6);
  unsafeAtomicAdd(op + lane, acc0 * nw);
  unsafeAtomicAdd(op + lane + 32, acc1 * nw);
}

// ---------------------------------------------------------------------------
extern "C" void kernel_launch(void* const* d_in, const int* in_sizes, int n_in,
                              void* d_out, int out_size, void* d_ws, size_t ws_size,
                              hipStream_t stream) {
  const float* feat   = (const float*)d_in[0];
  const float* norm   = (const float*)d_in[1];
  const float* weight = (const float*)d_in[2];
  const int*   etypes = (const int*)d_in[3];
  const int*   src    = (const int*)d_in[4];
  const int*   dst    = (const int*)d_in[5];
  float* out = (float*)d_out;

  const int N = in_sizes[0] / FEAT;   // 100000
  const int E = in_sizes[3];          // 1600000

  // Zero output every call (atomics accumulate; harness replays the graph).
  int n4 = out_size / 4;
  rgcn_zero_kernel<<<(n4 + 255) / 256, 256, 0, stream>>>((float4*)out, n4);

  const size_t slab = (size_t)N * FEAT * sizeof(float);  // 25.6 MB / relation
  int chunk = (int)(ws_size / slab);
  if (chunk > NUM_RELS) chunk = NUM_RELS;

  if (chunk >= 1 && (N % 16) == 0) {
    float* T = (float*)d_ws;
    for (int r0 = 0; r0 < NUM_RELS; r0 += chunk) {
      int r1 = r0 + chunk;
      if (r1 > NUM_RELS) r1 = NUM_RELS;
      dim3 grid(N / 16, r1 - r0);
      rgcn_gemm_wmma<<<grid, 128, 0, stream>>>(feat, weight, T, r0, N);
      long long tt = (long long)E * 16;
      int blocks = (int)((tt + 255) / 256);
      rgcn_edge_scatter<<<blocks, 256, 0, stream>>>(T, norm, etypes, src, dst,
                                                    out, r0, r1, N, E);
    }
  } else {
    int blocks = (E + 7) / 8;
    rgcn_edge_direct<<<blocks, 256, 0, stream>>>(feat, norm, etypes, src, dst,
                                                 weight, out, E);
  }
}